// FeatureGlobal_47193100648456
// MI455X (gfx1250) — compile-verified
//
#include <hip/hip_runtime.h>
#include <hip/hip_bf16.h>
#include <math.h>

typedef float v2f __attribute__((ext_vector_type(2)));
typedef float v8f __attribute__((ext_vector_type(8)));
typedef unsigned int u32x4 __attribute__((ext_vector_type(4)));
typedef int i32x4 __attribute__((ext_vector_type(4)));
typedef int i32x8 __attribute__((ext_vector_type(8)));

#define D_IN    128
#define HIDDEN  128
#define FEAT3   384
#define N_GRAPHS 50

// -------------------------------------------------------------------------
// zero fill
__global__ void zero_kernel(float* __restrict__ p, int n) {
    int i = blockIdx.x * blockDim.x + threadIdx.x;
    if (i < n) p[i] = 0.0f;
}

// -------------------------------------------------------------------------
// C[M,128] = X[M, :128 of lda] @ W[128,128], fp32 WMMA 16x16x4.
// 256 threads = 8 waves; W staged in 64KB LDS via the Tensor Data Mover
// (one wave-level DMA, TENSORcnt-tracked); each wave owns a 16x128 strip.
__global__ __launch_bounds__(256)
void gemm128_wmma_f32(const float* __restrict__ X, int lda,
                      const float* __restrict__ W,
                      float* __restrict__ C, int M) {
    __shared__ float sW[128 * 128];
    const int tid  = threadIdx.x;
    const int wave = tid >> 5;
    const int lane = tid & 31;
    const int rowBase = blockIdx.x * 128;

#if __has_builtin(__builtin_amdgcn_tensor_load_to_lds)
    if (wave == 0) {
        // D# descriptor (CDNA5 ISA):
        // 2D tensor 128x128 of 4B elements, full-tile load, row-major into sW.
        unsigned long long ga = (unsigned long long)(uintptr_t)W;
        unsigned int lds = (unsigned int)(uintptr_t)&sW[0]; // LDS aperture: addr[31:0]
        u32x4 g0;
        g0.x = 1u;                                      // count=1, user 2D descriptor
        g0.y = lds;                                     // lds_addr (bytes)
        g0.z = (unsigned int)ga;                        // global_addr[31:0]
        g0.w = (unsigned int)(ga >> 32) | 0x80000000u;  // addr[56:32] | type=2
        i32x8 g1;
        g1[0] = 0x20000;      // data_size=2 (4 bytes) in bits[17:16]
        g1[1] = 128 << 16;    // tensor_dim0 = 128   (bits 79:48)
        g1[2] = 128 << 16;    // tensor_dim1 = 128   (bits 111:80)
        g1[3] = 128 << 16;    // tile_dim0  = 128    (bits 127:112)
        g1[4] = 128;          // tile_dim1  = 128    (bits 143:128)
        g1[5] = 128;          // tensor_dim0_stride = 128 (bits 207:160, low32)
        g1[6] = 0x40000000;   // tensor_dim1_stride = 16384 (bits 255:208, low16<<16)
        g1[7] = 0;
        i32x4 z4 = {};
#if __clang_major__ >= 23
        i32x8 z8 = {};
        __builtin_amdgcn_tensor_load_to_lds(g0, g1, z4, z4, z8, 0);
#else
        __builtin_amdgcn_tensor_load_to_lds(g0, g1, z4, z4, 0);
#endif
        __builtin_amdgcn_s_wait_tensorcnt(0);           // s_wait_tensorcnt 0
    }
    __syncthreads();
#else
    for (int i = tid; i < (128 * 128) / 4; i += 256)
        ((float4*)sW)[i] = ((const float4*)W)[i];
    __syncthreads();
#endif

    const int tileRow = rowBase + wave * 16;
    if (tileRow >= M) return;                 // wave-uniform; EXEC stays all-1s

    const int half = lane >> 4;               // 0: K=0,1  1: K=2,3 (ISA striping)
    const int lm   = lane & 15;

    v8f acc[8] = {};
    const float* aRow = X + (size_t)(tileRow + lm) * lda + 2 * half;

    // pull this lane's 512B A-row into cache (global_prefetch_b8)
    __builtin_prefetch(aRow, 0, 3);
    __builtin_prefetch(aRow + 64, 0, 3);

    for (int k = 0; k < 128; k += 4) {
        // A frag 16x4: lane lm holds row, (v0,v1) = K=(k+2*half, k+1+2*half)
        v2f a = *(const v2f*)(aRow + k);
        const float* bp = &sW[(k + 2 * half) * 128 + lm];
#pragma unroll
        for (int j = 0; j < 8; ++j) {
            // B frag 4x16 of output cols [j*16, j*16+16)
            v2f b;
            b.x = bp[j * 16];
            b.y = bp[j * 16 + 128];
            acc[j] = __builtin_amdgcn_wmma_f32_16x16x4_f32(
                false, a, false, b, (short)0, acc[j], false, false);
        }
    }

    // C/D layout: VGPR r -> row (tileRow + r + 8*half), col = j*16 + lm
#pragma unroll
    for (int j = 0; j < 8; ++j) {
        float* cRow = C + (size_t)(tileRow + half * 8) * 128 + j * 16 + lm;
#pragma unroll
        for (int r = 0; r < 8; ++r)
            cRow[(size_t)r * 128] = acc[j][r];
    }
}

// -------------------------------------------------------------------------
// edge-parallel SpMM, 128 features: one wave per edge, 4 feats/lane.
// Dense operand (25.6MB) + accumulator (25.6MB) both fit in the 192MB L2,
// so the random gathers and the 102M f32 atomics resolve at L2 rate.
__global__ __launch_bounds__(256)
void spmm_edge128(const int* __restrict__ rows, const int* __restrict__ cols,
                  const float* __restrict__ vals, const float* __restrict__ Y,
                  float* __restrict__ acc, int nEdges) {
    int gid  = blockIdx.x * blockDim.x + threadIdx.x;
    int e    = gid >> 5;
    int lane = gid & 31;
    if (e >= nEdges) return;
    int   r = rows[e];
    int   c = cols[e];
    float v = vals[e];
    float4 y = *(const float4*)(Y + (size_t)c * 128 + lane * 4);
    float* o = acc + (size_t)r * 128 + lane * 4;
    atomicAdd(o + 0, v * y.x);
    atomicAdd(o + 1, v * y.y);
    atomicAdd(o + 2, v * y.z);
    atomicAdd(o + 3, v * y.w);
}

// -------------------------------------------------------------------------
// feat[:, colOff:colOff+128] = relu(acc + bias)
__global__ void bias_relu_to_feat(const float* __restrict__ acc,
                                  const float* __restrict__ bias,
                                  float* __restrict__ feat, int colOff, int N) {
    int idx = blockIdx.x * blockDim.x + threadIdx.x;
    if (idx >= N * 128) return;
    int row = idx >> 7, col = idx & 127;
    float v = acc[idx] + bias[col];
    v = v > 0.0f ? v : 0.0f;
    feat[(size_t)row * FEAT3 + colOff + col] = v;
}

// -------------------------------------------------------------------------
// s_pre[node] = dot(feat[node, :384], w_attn) ; wave-per-node, coalesced
__global__ __launch_bounds__(256)
void attn_dot(const float* __restrict__ feat, const float* __restrict__ w,
              float* __restrict__ sp, int N) {
    int gid  = blockIdx.x * blockDim.x + threadIdx.x;
    int node = gid >> 5;
    int lane = gid & 31;
    if (node >= N) return;
    const float* fr = feat + (size_t)node * FEAT3;
    float s = 0.0f;
    for (int i = lane; i < FEAT3; i += 32) s += fr[i] * w[i];
#pragma unroll
    for (int off = 16; off > 0; off >>= 1) s += __shfl_xor(s, off, 32);
    if (lane == 0) sp[node] = s;
}

// scalar SpMM for the attention score
__global__ void spmm_edge_scalar(const int* __restrict__ rows,
                                 const int* __restrict__ cols,
                                 const float* __restrict__ vals,
                                 const float* __restrict__ sp,
                                 float* __restrict__ sa, int E) {
    int e = blockIdx.x * blockDim.x + threadIdx.x;
    if (e >= E) return;
    atomicAdd(&sa[rows[e]], vals[e] * sp[cols[e]]);
}

__global__ void tanh_bias(const float* __restrict__ sa,
                          const float* __restrict__ b,
                          float* __restrict__ at, int N) {
    int i = blockIdx.x * blockDim.x + threadIdx.x;
    if (i < N) at[i] = tanhf(sa[i] + b[0]);
}

// -------------------------------------------------------------------------
// per-graph: rank-based stable top-k (matches lax.top_k tie-break), impor
// emission, then fused mean/max readout over the kept set (order-invariant).
__global__ __launch_bounds__(384)
void topk_readout(const float* __restrict__ feat, const float* __restrict__ attn,
                  float* __restrict__ readout, float* __restrict__ impor,
                  int n, int imporCap) {
    __shared__ float        ss[1024];
    __shared__ unsigned char km[1024];
    int g    = blockIdx.x;
    int tid  = threadIdx.x;
    int base = g * n;
    int keep = n >> 1;

    for (int i = tid; i < n; i += blockDim.x) ss[i] = attn[base + i];
    __syncthreads();

    for (int i = tid; i < n; i += blockDim.x) {
        float si   = ss[i];
        int   rank = 0;
        for (int j = 0; j < n; ++j) {
            float sj = ss[j];
            rank += ((sj > si) || ((sj == si) && (j < i))) ? 1 : 0;
        }
        km[i] = (rank < keep) ? 1 : 0;
        if (rank < keep) {
            int oi = g * keep + rank;           // local index, rank order
            if (oi < imporCap) impor[oi] = (float)i;
        }
    }
    __syncthreads();

    for (int f = tid; f < FEAT3; f += blockDim.x) {
        float sum = 0.0f, mx = -3.402823466e38f;
        for (int i = 0; i < n; ++i) {
            if (km[i]) {
                float v = feat[(size_t)(base + i) * FEAT3 + f] * ss[i];
                sum += v;
                if (v > mx) mx = v;
            }
        }
        readout[(size_t)g * (2 * FEAT3) + f]         = sum / (float)keep;
        readout[(size_t)g * (2 * FEAT3) + FEAT3 + f] = mx;
    }
}

// -------------------------------------------------------------------------
extern "C" void kernel_launch(void* const* d_in, const int* in_sizes, int n_in,
                              void* d_out, int out_size, void* d_ws, size_t ws_size,
                              hipStream_t stream) {
    const float* x        = (const float*)d_in[0];
    const float* adj_vals = (const float*)d_in[1];
    const float* Wmat[3]  = {(const float*)d_in[2], (const float*)d_in[4], (const float*)d_in[6]};
    const float* bvec[3]  = {(const float*)d_in[3], (const float*)d_in[5], (const float*)d_in[7]};
    const float* w_attn   = (const float*)d_in[8];
    const float* b_attn   = (const float*)d_in[9];
    const int*   adj_rows = (const int*)d_in[10];
    const int*   adj_cols = (const int*)d_in[11];

    const int N = in_sizes[0] / D_IN;    // 50000
    const int E = in_sizes[1];           // 800000
    const int nPer = N / N_GRAPHS;       // 1000
    float* out = (float*)d_out;

    // workspace carve-up (floats): A, B, feat, s_pre, s_acc, attn  (~129MB)
    float* A    = (float*)d_ws;
    float* B    = A    + (size_t)N * HIDDEN;
    float* feat = B    + (size_t)N * HIDDEN;
    float* sp   = feat + (size_t)N * FEAT3;
    float* sa   = sp + N;
    float* at   = sa + N;

    dim3 blk(256);
    int gemmGrid = (N + 127) / 128;
    int nmGrid   = (N * HIDDEN + 255) / 256;
    int edgeGrid = (E * 32 + 255) / 256;

    zero_kernel<<<(out_size + 255) / 256, blk, 0, stream>>>(out, out_size);

    const float* gin = x;
    int glda = D_IN;
    for (int layer = 0; layer < 3; ++layer) {
        gemm128_wmma_f32<<<gemmGrid, blk, 0, stream>>>(gin, glda, Wmat[layer], A, N);
        zero_kernel<<<nmGrid, blk, 0, stream>>>(B, N * HIDDEN);
        spmm_edge128<<<edgeGrid, blk, 0, stream>>>(adj_rows, adj_cols, adj_vals, A, B, E);
        bias_relu_to_feat<<<nmGrid, blk, 0, stream>>>(B, bvec[layer], feat, layer * HIDDEN, N);
        gin  = feat + layer * HIDDEN;
        glda = FEAT3;
    }

    attn_dot<<<(N * 32 + 255) / 256, blk, 0, stream>>>(feat, w_attn, sp, N);
    zero_kernel<<<(N + 255) / 256, blk, 0, stream>>>(sa, N);
    spmm_edge_scalar<<<(E + 255) / 256, blk, 0, stream>>>(adj_rows, adj_cols, adj_vals, sp, sa, E);
    tanh_bias<<<(N + 255) / 256, blk, 0, stream>>>(sa, b_attn, at, N);

    int imporCap = out_size - N_GRAPHS * 2 * FEAT3;
    topk_readout<<<N_GRAPHS, 384, 0, stream>>>(feat, at, out,
                                               out + N_GRAPHS * 2 * FEAT3,
                                               nPer, imporCap);
}